// RGCN_13013750907161
// MI455X (gfx1250) — compile-verified
//
#include <hip/hip_runtime.h>
#include <hip/hip_bf16.h>

typedef float v2f __attribute__((ext_vector_type(2)));
typedef float v8f __attribute__((ext_vector_type(8)));

#define NNODE 100000
#define NEDGE 640000
#define DMODEL 128
#define NEG_SLOPE 0.01f

// ---------------------------------------------------------------------------
// GEMM: Out[M,128] = A[M,128] @ Wm[128,128] + bias[128], fp32 via
// V_WMMA_F32_16X16X4_F32.  One wave per 16x16 output tile; 8 waves per block
// cover the full N=128; K=128 swept in 32 steps of 4.
// A-frag layout (ISA 16x4 fp32): lane r(0..15)/half h: VGPR0 = A[m0+r][k+2h],
// VGPR1 = A[m0+r][k+2h+1]  -> one float2 load per step.
// B-frag layout (4x16): VGPR0 = W[k+2h][n0+r], VGPR1 = W[k+2h+1][n0+r].
// C/D layout: VGPR i -> row m0+i+8h, col n0+r.
// ---------------------------------------------------------------------------
__global__ __launch_bounds__(256) void rgcn_gemm_wmma_f32(
    const float* __restrict__ A,
    const float* __restrict__ Wm,
    const float* __restrict__ bias,
    float* __restrict__ Out,
    int M)
{
    const int lane = threadIdx.x & 31;
    const int wave = threadIdx.x >> 5;   // 0..7 -> n tile
    const int r = lane & 15;
    const int h = lane >> 4;             // 0 or 1
    const int m0 = blockIdx.x * 16;
    const int n0 = wave * 16;
    if (m0 >= M) return;

    const float* arow = A + (size_t)(m0 + r) * DMODEL + 2 * h;
    const float* bcol = Wm + (size_t)(2 * h) * DMODEL + (n0 + r);

    v8f acc = {};
#pragma unroll
    for (int kk = 0; kk < 32; ++kk) {
        const int k = kk * 4;
        const float2 av = *reinterpret_cast<const float2*>(arow + k);
        v2f a; a.x = av.x; a.y = av.y;
        v2f b; b.x = bcol[(size_t)k * DMODEL];
               b.y = bcol[(size_t)(k + 1) * DMODEL];
        acc = __builtin_amdgcn_wmma_f32_16x16x4_f32(
            /*neg_a=*/false, a, /*neg_b=*/false, b,
            /*c_mod=*/(short)0, acc, /*reuse_a=*/false, /*reuse_b=*/false);
    }

    const float bb = bias[n0 + r];
#pragma unroll
    for (int i = 0; i < 8; ++i) {
        Out[(size_t)(m0 + i + 8 * h) * DMODEL + (n0 + r)] = acc[i] + bb;
    }
}

// ---------------------------------------------------------------------------
// Edge scatter with folded mean: agg[dst] += Wh[src] * invdeg[dst].
// One wave per edge; lane l moves a float4 (4 * global_atomic_add_f32).
// ---------------------------------------------------------------------------
__global__ __launch_bounds__(256) void rgcn_scatter_mean(
    const float* __restrict__ Wh,
    const int* __restrict__ src,
    const int* __restrict__ dst,
    const float* __restrict__ invdeg,
    float* __restrict__ agg,
    int n_edge)
{
    const int e = (int)((blockIdx.x * (unsigned)blockDim.x + threadIdx.x) >> 5);
    const int lane = threadIdx.x & 31;
    if (e >= n_edge) return;
    const int s = src[e];
    const int d = dst[e];
    const float sc = invdeg[d];
    const float4 v = *reinterpret_cast<const float4*>(Wh + (size_t)s * DMODEL + lane * 4);
    float* base = agg + (size_t)d * DMODEL + lane * 4;
    atomicAdd(base + 0, v.x * sc);
    atomicAdd(base + 1, v.y * sc);
    atomicAdd(base + 2, v.z * sc);
    atomicAdd(base + 3, v.w * sc);
}

// ---------------------------------------------------------------------------
__global__ __launch_bounds__(256) void rgcn_zero4(float4* __restrict__ p, int n4)
{
    const int i = blockIdx.x * blockDim.x + threadIdx.x;
    if (i < n4) p[i] = make_float4(0.f, 0.f, 0.f, 0.f);
}

__global__ __launch_bounds__(256) void rgcn_degree(
    const int* __restrict__ dst, float* __restrict__ deg, int n_edge)
{
    const int i = blockIdx.x * blockDim.x + threadIdx.x;
    if (i < n_edge) atomicAdd(&deg[dst[i]], 1.0f);
}

__global__ __launch_bounds__(256) void rgcn_invdeg(float* __restrict__ deg, int n)
{
    const int i = blockIdx.x * blockDim.x + threadIdx.x;
    if (i < n) deg[i] = 1.0f / fmaxf(deg[i], 1.0f);
}

__global__ __launch_bounds__(256) void rgcn_lrelu4(
    const float4* __restrict__ agg, float4* __restrict__ out, int n4)
{
    const int i = blockIdx.x * blockDim.x + threadIdx.x;
    if (i < n4) {
        float4 v = agg[i];
        v.x = v.x > 0.f ? v.x : NEG_SLOPE * v.x;
        v.y = v.y > 0.f ? v.y : NEG_SLOPE * v.y;
        v.z = v.z > 0.f ? v.z : NEG_SLOPE * v.z;
        v.w = v.w > 0.f ? v.w : NEG_SLOPE * v.w;
        out[i] = v;
    }
}

// ---------------------------------------------------------------------------
extern "C" void kernel_launch(void* const* d_in, const int* in_sizes, int n_in,
                              void* d_out, int out_size, void* d_ws, size_t ws_size,
                              hipStream_t stream)
{
    const float* h_user = (const float*)d_in[0];
    const float* h_item = (const float*)d_in[1];
    const float* W      = (const float*)d_in[2];   // [3,3,128,128]
    const float* b      = (const float*)d_in[3];   // [3,3,128]
    const int* f_src    = (const int*)d_in[4];
    const int* f_dst    = (const int*)d_in[5];
    const int* c_src    = (const int*)d_in[6];
    const int* c_dst    = (const int*)d_in[7];
    const int* cb_src   = (const int*)d_in[8];
    const int* cb_dst   = (const int*)d_in[9];

    const size_t ND = (size_t)NNODE * DMODEL;      // 12.8M floats
    float* ws      = (float*)d_ws;
    float* hu_buf  = ws;                 // [N,128]
    float* hi_buf  = ws + ND;            // [N,128]
    float* Wh      = ws + 2 * ND;        // [N,128] (reused across etypes)
    float* agg_u   = ws + 3 * ND;        // [N,128]
    float* agg_i   = ws + 4 * ND;        // [N,128]
    float* idg_f   = ws + 5 * ND;        // [N]
    float* idg_c   = idg_f + NNODE;
    float* idg_cb  = idg_c + NNODE;

    const dim3 blk(256);
    const int nd4_blocks   = (int)((ND / 4 + 255) / 256);          // agg/h zero & lrelu
    const int deg4_blocks  = (3 * NNODE / 4 + 255) / 256;
    const int edge_blocks  = (NEDGE + 255) / 256;                  // thread-per-edge
    const int scat_blocks  = (NEDGE + 7) / 8;                      // wave-per-edge
    const int node_blocks  = (3 * NNODE + 255) / 256;
    const int gemm_blocks  = NNODE / 16;                           // 6250, exact

    // --- degrees -> inverse degrees (per call; deterministic) ---
    rgcn_zero4<<<deg4_blocks, blk, 0, stream>>>((float4*)idg_f, 3 * NNODE / 4);
    rgcn_degree<<<edge_blocks, blk, 0, stream>>>(f_dst,  idg_f,  NEDGE);
    rgcn_degree<<<edge_blocks, blk, 0, stream>>>(c_dst,  idg_c,  NEDGE);
    rgcn_degree<<<edge_blocks, blk, 0, stream>>>(cb_dst, idg_cb, NEDGE);
    rgcn_invdeg<<<node_blocks, blk, 0, stream>>>(idg_f, 3 * NNODE);

    const float* hu = h_user;
    const float* hi = h_item;

    for (int l = 0; l < 3; ++l) {
        const float* Wl = W + (size_t)l * 3 * DMODEL * DMODEL;
        const float* bl = b + (size_t)l * 3 * DMODEL;

        rgcn_zero4<<<nd4_blocks, blk, 0, stream>>>((float4*)agg_u, (int)(ND / 4));
        rgcn_zero4<<<nd4_blocks, blk, 0, stream>>>((float4*)agg_i, (int)(ND / 4));

        // etype 0: user --follows--> user
        rgcn_gemm_wmma_f32<<<gemm_blocks, blk, 0, stream>>>(hu, Wl, bl, Wh, NNODE);
        rgcn_scatter_mean<<<scat_blocks, blk, 0, stream>>>(Wh, f_src, f_dst, idg_f, agg_u, NEDGE);

        // etype 1: user --clicks--> item
        rgcn_gemm_wmma_f32<<<gemm_blocks, blk, 0, stream>>>(
            hu, Wl + DMODEL * DMODEL, bl + DMODEL, Wh, NNODE);
        rgcn_scatter_mean<<<scat_blocks, blk, 0, stream>>>(Wh, c_src, c_dst, idg_c, agg_i, NEDGE);

        // etype 2: item --clicked_by--> user
        rgcn_gemm_wmma_f32<<<gemm_blocks, blk, 0, stream>>>(
            hi, Wl + 2 * DMODEL * DMODEL, bl + 2 * DMODEL, Wh, NNODE);
        rgcn_scatter_mean<<<scat_blocks, blk, 0, stream>>>(Wh, cb_src, cb_dst, idg_cb, agg_u, NEDGE);

        float* out_u = (l == 2) ? (float*)d_out       : hu_buf;
        float* out_i = (l == 2) ? (float*)d_out + ND  : hi_buf;
        rgcn_lrelu4<<<nd4_blocks, blk, 0, stream>>>((const float4*)agg_u, (float4*)out_u, (int)(ND / 4));
        rgcn_lrelu4<<<nd4_blocks, blk, 0, stream>>>((const float4*)agg_i, (float4*)out_i, (int)(ND / 4));

        hu = hu_buf;
        hi = hi_buf;
    }
}